// GraphTransformer_17428977287426
// MI455X (gfx1250) — compile-verified
//
#include <hip/hip_runtime.h>
#include <hip/hip_bf16.h>
#include <math.h>

#define NNODES  100000
#define NEDGES  800000
#define FEAT    64
#define HEADS   4
#define HDIM    32
#define HIDDEN  128
#define NGRAPH  512

typedef __attribute__((ext_vector_type(2))) float v2f;
typedef __attribute__((ext_vector_type(4))) float v4f;
typedef __attribute__((ext_vector_type(8))) float v8f;

// ---------- order-preserving float<->int for atomicMax segment-max ----------
__device__ __forceinline__ int f32_to_ordered(float f) {
  int i = __float_as_int(f);
  return i >= 0 ? i : (i ^ 0x7fffffff);
}
__device__ __forceinline__ float ordered_to_f32(int i) {
  return __int_as_float(i >= 0 ? i : (i ^ 0x7fffffff));
}

// ---------------------------------------------------------------------------
// Y[M,128] = X[M,K] @ W[K,128] + b      (M % 16 == 0, K % 4 == 0, K static)
// One wave per 16x16 output tile, fp32 WMMA 16x16x4, fully unrolled K chain.
// A layout (16x4 f32): lanes 0-15 rows M=0..15, VGPR0/1 = K {0,1}; lanes 16-31
// same rows, K {2,3}.  B layout symmetric (rows striped across lanes).
// ---------------------------------------------------------------------------
template <int K>
__global__ __launch_bounds__(256) void gemm_bias_kernel(
    const float* __restrict__ X, const float* __restrict__ W,
    const float* __restrict__ bias, float* __restrict__ Y) {
  const int lane = threadIdx.x & 31;
  const int wave = threadIdx.x >> 5;        // 0..7 -> column tile (8*16 = 128)
  const int mlo  = lane & 15;
  const int hi   = lane >> 4;               // 0 or 1
  const int row  = blockIdx.x * 16 + mlo;   // A row for this lane
  const int col  = wave * 16 + mlo;         // B/C/D column for this lane

  const float* __restrict__ Xr = X + (size_t)row * K + hi * 2;
  const float* __restrict__ Wc = W + (size_t)(hi * 2) * HIDDEN + col;

  v8f acc = {};
#pragma unroll
  for (int kk = 0; kk < K / 4; ++kk) {
    v2f a = *(const v2f*)(Xr + kk * 4);     // K even offsets -> 8B aligned
    v2f b;
    b.x = Wc[(kk * 4 + 0) * HIDDEN];
    b.y = Wc[(kk * 4 + 1) * HIDDEN];
    acc = __builtin_amdgcn_wmma_f32_16x16x4_f32(
        /*neg_a=*/false, a, /*neg_b=*/false, b,
        /*c_mod=*/(short)0, acc, /*reuse_a=*/false, /*reuse_b=*/false);
  }
  const float bv = bias[col];
  const int rbase = blockIdx.x * 16 + hi * 8;          // C/D: vgpr i -> row i(+8)
#pragma unroll
  for (int i = 0; i < 8; ++i)
    Y[(size_t)(rbase + i) * HIDDEN + col] = acc[i] + bv;
}

// ---------------------------------------------------------------------------
__global__ void fill_int_kernel(int* __restrict__ p, int v, int n) {
  int i = blockIdx.x * blockDim.x + threadIdx.x;
  if (i < n) p[i] = v;
}

// scores[e,h] = <q[dst,h,:], k[src,h,:]> / sqrt(D); segment-max into mred[dst,h]
__global__ void edge_score_kernel(const float* __restrict__ q,
                                  const float* __restrict__ k,
                                  const int* __restrict__ src,
                                  const int* __restrict__ dst,
                                  float* __restrict__ scores,
                                  int* __restrict__ mred) {
  int tid = blockIdx.x * blockDim.x + threadIdx.x;
  if (tid >= NEDGES * HEADS) return;
  const int e = tid >> 2, h = tid & 3;
  const int s = src[e], d = dst[e];
  const v4f* qr = (const v4f*)(q + (size_t)d * HIDDEN + h * HDIM);
  const v4f* kr = (const v4f*)(k + (size_t)s * HIDDEN + h * HDIM);
  float acc = 0.f;
#pragma unroll
  for (int i = 0; i < HDIM / 4; ++i) {
    v4f a = qr[i], b = kr[i];
    acc += a.x * b.x + a.y * b.y + a.z * b.z + a.w * b.w;
  }
  acc *= 0.1767766952966369f;  // 1/sqrt(32)
  scores[tid] = acc;
  atomicMax(&mred[d * HEADS + h], f32_to_ordered(acc));
}

// ex = exp(score - max); denom[dst,h] += ex; agg[dst,h,:] += ex * v[src,h,:]
__global__ void edge_scatter_kernel(const float* __restrict__ vmat,
                                    const float* __restrict__ scores,
                                    const int* __restrict__ src,
                                    const int* __restrict__ dst,
                                    const int* __restrict__ mred,
                                    float* __restrict__ denom,
                                    float* __restrict__ agg) {
  int tid = blockIdx.x * blockDim.x + threadIdx.x;
  if (tid >= NEDGES * HEADS) return;
  const int e = tid >> 2, h = tid & 3;
  const int s = src[e], d = dst[e];
  const float mm = ordered_to_f32(mred[d * HEADS + h]);
  const float ex = __expf(scores[tid] - mm);
  atomicAdd(&denom[d * HEADS + h], ex);
  const v4f* vr = (const v4f*)(vmat + (size_t)s * HIDDEN + h * HDIM);
  float* ar = agg + (size_t)d * HIDDEN + h * HDIM;
#pragma unroll
  for (int i = 0; i < HDIM / 4; ++i) {
    v4f vv = vr[i];
    atomicAdd(&ar[4 * i + 0], ex * vv.x);
    atomicAdd(&ar[4 * i + 1], ex * vv.y);
    atomicAdd(&ar[4 * i + 2], ex * vv.z);
    atomicAdd(&ar[4 * i + 3], ex * vv.w);
  }
}

// out = relu(agg/denom + skip); empty segments -> skip only (agg contributes 0)
__global__ void epilogue_relu_kernel(const float* __restrict__ agg,
                                     const float* __restrict__ denom,
                                     const float* __restrict__ skip,
                                     float* __restrict__ out) {
  int tid = blockIdx.x * blockDim.x + threadIdx.x;
  if (tid >= NNODES * HIDDEN) return;
  const int n = tid >> 7, c = tid & 127, h = c >> 5;
  const float dn = denom[n * HEADS + h];
  const float a = dn > 0.f ? agg[tid] / dn : 0.f;
  out[tid] = fmaxf(a + skip[tid], 0.f);
}

// pooled[g,:] += h[n,:]; cnt[g] += 1
__global__ void pool_scatter_kernel(const float* __restrict__ hmat,
                                    const int* __restrict__ batch,
                                    float* __restrict__ pooled,
                                    float* __restrict__ cnt) {
  int tid = blockIdx.x * blockDim.x + threadIdx.x;
  if (tid >= NNODES * HIDDEN) return;
  const int n = tid >> 7, c = tid & 127;
  const int g = batch[n];
  atomicAdd(&pooled[(size_t)g * HIDDEN + c], hmat[tid]);
  if (c == 0) atomicAdd(&cnt[g], 1.f);
}

// per-graph mean + two 128->1 heads
__global__ __launch_bounds__(128) void head_kernel(
    const float* __restrict__ pooled, const float* __restrict__ cnt,
    const float* __restrict__ Wmem, const float* __restrict__ bmem,
    const float* __restrict__ Wtime, const float* __restrict__ btime,
    float* __restrict__ out) {
  __shared__ float sm[HIDDEN];
  __shared__ float st[HIDDEN];
  const int g = blockIdx.x, t = threadIdx.x;
  const float c = fmaxf(cnt[g], 1.f);
  const float pv = pooled[(size_t)g * HIDDEN + t] / c;
  sm[t] = pv * Wmem[t];
  st[t] = pv * Wtime[t];
  __syncthreads();
  for (int off = 64; off > 0; off >>= 1) {
    if (t < off) { sm[t] += sm[t + off]; st[t] += st[t + off]; }
    __syncthreads();
  }
  if (t == 0) {
    out[g] = sm[0] + bmem[0];
    out[NGRAPH + g] = st[0] + btime[0];
  }
}

// ---------------------------------------------------------------------------
extern "C" void kernel_launch(void* const* d_in, const int* in_sizes, int n_in,
                              void* d_out, int out_size, void* d_ws, size_t ws_size,
                              hipStream_t stream) {
  (void)in_sizes; (void)n_in; (void)out_size; (void)ws_size;
  const float* x    = (const float*)d_in[0];
  const int* eidx   = (const int*)d_in[1];
  const int* batch  = (const int*)d_in[2];
  const float* Wq1 = (const float*)d_in[3],  *bq1 = (const float*)d_in[4];
  const float* Wk1 = (const float*)d_in[5],  *bk1 = (const float*)d_in[6];
  const float* Wv1 = (const float*)d_in[7],  *bv1 = (const float*)d_in[8];
  const float* Ws1 = (const float*)d_in[9],  *bs1 = (const float*)d_in[10];
  const float* Wq2 = (const float*)d_in[11], *bq2 = (const float*)d_in[12];
  const float* Wk2 = (const float*)d_in[13], *bk2 = (const float*)d_in[14];
  const float* Wv2 = (const float*)d_in[15], *bv2 = (const float*)d_in[16];
  const float* Ws2 = (const float*)d_in[17], *bs2 = (const float*)d_in[18];
  const float* Wmem = (const float*)d_in[19], *bmem = (const float*)d_in[20];
  const float* Wtim = (const float*)d_in[21], *btim = (const float*)d_in[22];
  float* out = (float*)d_out;

  const int* srcIdx = eidx;           // edge_index[0]
  const int* dstIdx = eidx + NEDGES;  // edge_index[1]

  // ---- workspace layout ----
  const size_t NH = (size_t)NNODES * HIDDEN;
  float* q      = (float*)d_ws;
  float* k      = q + NH;
  float* v      = k + NH;
  float* s      = v + NH;
  float* agg    = s + NH;
  float* h1     = agg + NH;
  float* h2     = h1 + NH;
  float* scores = h2 + NH;                         // E*H
  float* denom  = scores + (size_t)NEDGES * HEADS; // N*H
  int*   mred   = (int*)(denom + (size_t)NNODES * HEADS);
  float* pooled = (float*)(mred + (size_t)NNODES * HEADS); // G*HID
  float* cnt    = pooled + (size_t)NGRAPH * HIDDEN;        // G

  const int mTiles   = NNODES / 16;                        // 6250
  const int eBlocks  = (NEDGES * HEADS + 255) / 256;
  const int nhBlocks = (NNODES * HIDDEN + 255) / 256;

  // ================= Layer 1 (K = 64) =================
  gemm_bias_kernel<FEAT><<<mTiles, 256, 0, stream>>>(x, Wq1, bq1, q);
  gemm_bias_kernel<FEAT><<<mTiles, 256, 0, stream>>>(x, Wk1, bk1, k);
  gemm_bias_kernel<FEAT><<<mTiles, 256, 0, stream>>>(x, Wv1, bv1, v);
  gemm_bias_kernel<FEAT><<<mTiles, 256, 0, stream>>>(x, Ws1, bs1, s);

  hipMemsetAsync(agg, 0, NH * sizeof(float), stream);
  hipMemsetAsync(denom, 0, (size_t)NNODES * HEADS * sizeof(float), stream);
  fill_int_kernel<<<(NNODES * HEADS + 255) / 256, 256, 0, stream>>>(
      mred, INT_MIN, NNODES * HEADS);

  edge_score_kernel<<<eBlocks, 256, 0, stream>>>(q, k, srcIdx, dstIdx, scores, mred);
  edge_scatter_kernel<<<eBlocks, 256, 0, stream>>>(v, scores, srcIdx, dstIdx,
                                                   mred, denom, agg);
  epilogue_relu_kernel<<<nhBlocks, 256, 0, stream>>>(agg, denom, s, h1);

  // ================= Layer 2 (K = 128) =================
  gemm_bias_kernel<HIDDEN><<<mTiles, 256, 0, stream>>>(h1, Wq2, bq2, q);
  gemm_bias_kernel<HIDDEN><<<mTiles, 256, 0, stream>>>(h1, Wk2, bk2, k);
  gemm_bias_kernel<HIDDEN><<<mTiles, 256, 0, stream>>>(h1, Wv2, bv2, v);
  gemm_bias_kernel<HIDDEN><<<mTiles, 256, 0, stream>>>(h1, Ws2, bs2, s);

  hipMemsetAsync(agg, 0, NH * sizeof(float), stream);
  hipMemsetAsync(denom, 0, (size_t)NNODES * HEADS * sizeof(float), stream);
  fill_int_kernel<<<(NNODES * HEADS + 255) / 256, 256, 0, stream>>>(
      mred, INT_MIN, NNODES * HEADS);

  edge_score_kernel<<<eBlocks, 256, 0, stream>>>(q, k, srcIdx, dstIdx, scores, mred);
  edge_scatter_kernel<<<eBlocks, 256, 0, stream>>>(v, scores, srcIdx, dstIdx,
                                                   mred, denom, agg);
  epilogue_relu_kernel<<<nhBlocks, 256, 0, stream>>>(agg, denom, s, h2);

  // ================= Pool + heads =================
  hipMemsetAsync(pooled, 0, (size_t)NGRAPH * HIDDEN * sizeof(float), stream);
  hipMemsetAsync(cnt, 0, (size_t)NGRAPH * sizeof(float), stream);
  pool_scatter_kernel<<<nhBlocks, 256, 0, stream>>>(h2, batch, pooled, cnt);
  head_kernel<<<NGRAPH, 128, 0, stream>>>(pooled, cnt, Wmem, bmem, Wtim, btim, out);
}